// AdaptiveSpecAugment_56461640073510
// MI455X (gfx1250) — compile-verified
//
#include <hip/hip_runtime.h>
#include <stdint.h>

#ifndef __has_builtin
#define __has_builtin(x) 0
#endif

#if __has_builtin(__builtin_amdgcn_global_load_async_to_lds_b128) && \
    __has_builtin(__builtin_amdgcn_s_wait_asynccnt)
#define USE_ASYNC 1
#else
#define USE_ASYNC 0
#endif

/* Compile-only loop: force a hard failure if the device pass did NOT get the
   CDNA5 async-to-LDS builtins, so a successful compile proves the async path
   is in the emitted gfx1250 code. (Host pass has no amdgcn builtins — skip.) */
#if defined(__HIP_DEVICE_COMPILE__) && !USE_ASYNC
#error "gfx1250 async-to-LDS builtins not available in device compile"
#endif

typedef float f4 __attribute__((ext_vector_type(4)));
typedef int   i4 __attribute__((ext_vector_type(4)));

#define B_DIM 256
#define T_DIM 1600
#define F_DIM 80
#define N_ELEM (B_DIM * T_DIM * F_DIM)   /* 32,768,000 */
#define N_F4   (N_ELEM / 4)              /* 8,192,000  */
#define TPB    256
#define NBLK   4000
#define NITER  8                          /* TPB*NBLK*NITER == N_F4 exactly */

/* ---------------- threefry2x32: exact JAX PRNG block cipher ---------------- */
__device__ __forceinline__ uint32_t rotl32(uint32_t v, int n) {
  return (v << n) | (v >> (32 - n));
}

__device__ void threefry2x32(uint32_t k0, uint32_t k1,
                             uint32_t x0, uint32_t x1,
                             uint32_t& o0, uint32_t& o1) {
  const uint32_t ks2 = k0 ^ k1 ^ 0x1BD11BDAu;
  x0 += k0; x1 += k1;
#define TF_ROUND(r) { x0 += x1; x1 = rotl32(x1, r); x1 ^= x0; }
  TF_ROUND(13) TF_ROUND(15) TF_ROUND(26) TF_ROUND(6)
  x0 += k1;  x1 += ks2 + 1u;
  TF_ROUND(17) TF_ROUND(29) TF_ROUND(16) TF_ROUND(24)
  x0 += ks2; x1 += k0 + 2u;
  TF_ROUND(13) TF_ROUND(15) TF_ROUND(26) TF_ROUND(6)
  x0 += k0;  x1 += k1 + 3u;
  TF_ROUND(17) TF_ROUND(29) TF_ROUND(16) TF_ROUND(24)
  x0 += k1;  x1 += ks2 + 4u;
  TF_ROUND(13) TF_ROUND(15) TF_ROUND(26) TF_ROUND(6)
  x0 += ks2; x1 += k0 + 5u;
#undef TF_ROUND
  o0 = x0; o1 = x1;
}

__device__ __forceinline__ float bits_to_unif(uint32_t b) {
  /* JAX uniform: bitcast((bits>>9)|0x3f800000, f32) - 1.0  in [0,1) */
  return __uint_as_float((b >> 9) | 0x3F800000u) - 1.0f;
}

/* --------- mask generation: per-batch freq bitmask + per-(b,t) byte -------- */
__global__ __launch_bounds__(256)
void specaug_mask_kernel(const int* __restrict__ length,
                         uint32_t* __restrict__ fmask,   /* [B][4] words, bits over F */
                         uint8_t*  __restrict__ tmask,   /* [B][T] bytes              */
                         int* __restrict__ out_tail, int write_tail) {
  const int b = blockIdx.x;
  __shared__ int s_yl[10], s_wt[10], s_nt;

  if (threadIdx.x == 0) {
    /* split(key(42), 4): cipher (0,42) on counts (i, i+4) -> a_i (out0), b_i (out1)
       keys: kf1=(a0,a1) kf2=(a2,a3) kt1=(b0,b1) kt2=(b2,b3) */
    uint32_t a0,a1,a2,a3,c0,c1,c2,c3;
    threefry2x32(0u, 42u, 0u, 4u, a0, c0);
    threefry2x32(0u, 42u, 1u, 5u, a1, c1);
    threefry2x32(0u, 42u, 2u, 6u, a2, c2);
    threefry2x32(0u, 42u, 3u, 7u, a3, c3);

    /* frequency masks: randint spans 54 (x_left) and 28 (width) over (B,2);
       bits shape (2,B,2): element (b,m) -> cipher count pair (j, j+512), j=2b+m;
       hi=out0, lo=out1 */
    const uint32_t SPAN_X = (uint32_t)(F_DIM - 27 + 1);                       /* 54 */
    const uint32_t SPAN_W = 28u;
    const uint32_t MULT_X = ((65536u % SPAN_X) * (65536u % SPAN_X)) % SPAN_X; /* 22 */
    const uint32_t MULT_W = ((65536u % SPAN_W) * (65536u % SPAN_W)) % SPAN_W; /*  4 */

    uint32_t xl[2], wf[2];
    for (int m = 0; m < 2; ++m) {
      const uint32_t j = (uint32_t)(b * 2 + m);
      uint32_t hi, lo;
      threefry2x32(a0, a1, j, j + 512u, hi, lo);
      xl[m] = ((hi % SPAN_X) * MULT_X + (lo % SPAN_X)) % SPAN_X;
      threefry2x32(a2, a3, j, j + 512u, hi, lo);
      wf[m] = ((hi % SPAN_W) * MULT_W + (lo % SPAN_W)) % SPAN_W;
    }
    uint32_t fm0 = 0u, fm1 = 0u, fm2 = 0u;
    for (int f = 0; f < F_DIM; ++f) {
      bool msk = false;
      for (int m = 0; m < 2; ++m)
        msk = msk || ((uint32_t)f >= xl[m] && (uint32_t)f < xl[m] + wf[m]);
      if (msk) {
        if (f < 32)      fm0 |= 1u << f;
        else if (f < 64) fm1 |= 1u << (f - 32);
        else             fm2 |= 1u << (f - 64);
      }
    }
    fmask[b * 4 + 0] = fm0; fmask[b * 4 + 1] = fm1;
    fmask[b * 4 + 2] = fm2; fmask[b * 4 + 3] = 0u;

    /* time masks: uniforms shape (B,10) -> flat j=10b+k, half=1280:
       j<1280: out0 of cipher (j, j+1280); else out1 of cipher (j-1280, j) */
    const int L  = length[b];
    const float lf = (float)L;
    const int nt  = min(10, (int)(lf * 0.05f));
    const int tw  = max(1,  (int)(lf * 0.05f));
    const int hiv = max(1, L - tw);
    s_nt = nt;
    for (int k = 0; k < 10; ++k) {
      const int j = b * 10 + k;
      uint32_t o0, o1, bits1, bits2;
      if (j < 1280) {
        threefry2x32(c0, c1, (uint32_t)j, (uint32_t)(j + 1280), o0, o1); bits1 = o0;
        threefry2x32(c2, c3, (uint32_t)j, (uint32_t)(j + 1280), o0, o1); bits2 = o0;
      } else {
        threefry2x32(c0, c1, (uint32_t)(j - 1280), (uint32_t)j, o0, o1); bits1 = o1;
        threefry2x32(c2, c3, (uint32_t)(j - 1280), (uint32_t)j, o0, o1); bits2 = o1;
      }
      s_yl[k] = min((int)(bits_to_unif(bits1) * (float)(hiv + 1)), hiv);
      s_wt[k] = min((int)(bits_to_unif(bits2) * (float)(tw  + 1)), tw);
    }
    if (write_tail) out_tail[b] = L;   /* length passthrough (int bits in d_out tail) */
  }
  __syncthreads();

  const int nt = s_nt;
  for (int t = threadIdx.x; t < T_DIM; t += blockDim.x) {
    int m = 0;
#pragma unroll
    for (int k = 0; k < 10; ++k)
      m |= (int)((k < nt) & (t >= s_yl[k]) & (t < s_yl[k] + s_wt[k]));
    tmask[b * T_DIM + t] = (uint8_t)m;
  }
}

/* ---------- bulk stream: async-to-LDS double buffer, B128, NT stores ------- */
__device__ __forceinline__ void cp_async16(const f4* gsrc, f4* ldst) {
#if USE_ASYNC
  __attribute__((address_space(1))) i4* gp =
      (__attribute__((address_space(1))) i4*)(unsigned long long)(uintptr_t)gsrc;
  __attribute__((address_space(3))) i4* lp =
      (__attribute__((address_space(3))) i4*)(unsigned int)(uintptr_t)ldst;
  __builtin_amdgcn_global_load_async_to_lds_b128(gp, lp, 0, 0);
#else
  (void)gsrc; (void)ldst;
#endif
}

__global__ __launch_bounds__(TPB)
void specaug_apply_kernel(const float* __restrict__ in, float* __restrict__ out,
                          const uint8_t* __restrict__ tmask,
                          const uint32_t* __restrict__ fmask) {
  const int tid    = threadIdx.x;
  const int base   = blockIdx.x * TPB + tid;     /* < 1,024,000 */
  const int stride = TPB * NBLK;                 /* 1,024,000   */
  const f4* __restrict__ in4 = (const f4*)in;
  f4* __restrict__ out4      = (f4*)out;

#if USE_ASYNC
  __shared__ f4 sbuf[2][TPB];                    /* 8 KB of the 320 KB LDS */
  cp_async16(in4 + base, &sbuf[0][tid]);         /* prologue */
#endif

#pragma unroll
  for (int i = 0; i < NITER; ++i) {
    const int g = base + i * stride;             /* float4 index, fully coalesced */
#if USE_ASYNC
    if (i + 1 < NITER) {
      cp_async16(in4 + (g + stride), &sbuf[(i + 1) & 1][tid]);
      __builtin_amdgcn_s_wait_asynccnt(1);       /* oldest (iter i) landed in LDS */
    } else {
      __builtin_amdgcn_s_wait_asynccnt(0);
    }
    asm volatile("" ::: "memory");
    const f4 v = sbuf[i & 1][tid];
#else
    const f4 v = in4[g];
#endif
    const int row = g / 20;                      /* (b,t) row: 20 float4 per row */
    const int c4  = g - row * 20;                /* float4 within row            */
    const int bb  = row / T_DIM;                 /* batch                        */
    const uint32_t tm  = tmask[row];
    const uint32_t fmw = fmask[(bb << 2) + (c4 >> 3)];
    const uint32_t nib = (fmw >> ((c4 & 7) << 2)) & 0xFu;
    f4 r;
    r.x = ((tm | ( nib       & 1u)) != 0u) ? 0.0f : v.x;
    r.y = ((tm | ((nib >> 1) & 1u)) != 0u) ? 0.0f : v.y;
    r.z = ((tm | ((nib >> 2) & 1u)) != 0u) ? 0.0f : v.z;
    r.w = ((tm | ((nib >> 3) & 1u)) != 0u) ? 0.0f : v.w;
    __builtin_nontemporal_store(r, out4 + g);    /* stream out, don't thrash L2 */
  }
}

/* --------------------------------- launch --------------------------------- */
extern "C" void kernel_launch(void* const* d_in, const int* in_sizes, int n_in,
                              void* d_out, int out_size, void* d_ws, size_t ws_size,
                              hipStream_t stream) {
  (void)in_sizes; (void)n_in; (void)ws_size;
  const float* in     = (const float*)d_in[0];
  const int*   length = (const int*)d_in[1];
  float*       out    = (float*)d_out;

  uint32_t* fmask = (uint32_t*)d_ws;                     /* 4 KB   */
  uint8_t*  tmask = (uint8_t*)d_ws + 4096;               /* 400 KB */
  const int write_tail = (out_size >= N_ELEM + B_DIM) ? 1 : 0;

  specaug_mask_kernel<<<B_DIM, 256, 0, stream>>>(length, fmask, tmask,
                                                 (int*)d_out + N_ELEM, write_tail);
  specaug_apply_kernel<<<NBLK, TPB, 0, stream>>>(in, out, tmask, fmask);
}